// STU_61701500174893
// MI455X (gfx1250) — compile-verified
//
#include <hip/hip_runtime.h>

// STU for MI455X / gfx1250 (wave32).
// K1: fused causal-Toeplitz conv + eig^{1/4} + phi projection + AR, fp32 WMMA.
//     Paired time tiles {bx, 63-bx} for uniform per-block work; filter column
//     staged to LDS with global_load_async_to_lds_b32 (ASYNCcnt path).
// K2: order-2 matrix recurrence; m_y 3/4 register-resident (WGP regfile = 512 KB
//     = |m_y|), 2 threads per channel, shfl-pair reduction.

#define B_  8
#define L_  1024
#define D_  256
#define K_  24
#define KU_ 3
#define KY_ 2
#define NT_ (L_ / 16)  // 64 time tiles

typedef __attribute__((ext_vector_type(2))) float v2f;
typedef __attribute__((ext_vector_type(8))) float v8f;

__device__ __forceinline__ v8f wmma_f32(v2f a, v2f b, v8f c) {
  // (neg_a, A, neg_b, B, c_mod, C, reuse_a, reuse_b)
  return __builtin_amdgcn_wmma_f32_16x16x4_f32(false, a, false, b, (short)0, c,
                                               false, false);
}

typedef __attribute__((address_space(3))) const void lds_cv;
__device__ __forceinline__ unsigned lds_addr32(const void* p) {
  return (unsigned)(uintptr_t)(lds_cv*)p;  // generic -> LDS byte offset
}

__device__ __forceinline__ void async_f32_to_lds(unsigned lds_byte,
                                                 const float* g) {
  asm volatile("global_load_async_to_lds_b32 %0, %1, off"
               :: "v"(lds_byte), "v"(g) : "memory");
}
__device__ __forceinline__ void wait_async0() {
  asm volatile("s_wait_asynccnt 0x0" ::: "memory");
}

// ---------------------------------------------------------------------------
// Kernel 1: delta = x_tilde @ m_phi + AR(m_u, u)
// grid = (NT_/2, B), block = 256 (8 waves); wave w owns cols [32w, 32w+32)
// ---------------------------------------------------------------------------
__global__ __launch_bounds__(256) void stu_delta_kernel(
    const float* __restrict__ inp,       // [B, L, D]
    const float* __restrict__ m_u,       // [D, D, KU]
    const float* __restrict__ m_phi,     // [K*D, D]
    const float* __restrict__ eig_vals,  // [K]
    const float* __restrict__ eig_vecs,  // [L, K]
    float* __restrict__ delta) {         // [B, L, D]
  const int bx   = blockIdx.x;
  const int b    = blockIdx.y;
  const int lane = threadIdx.x & 31;
  const int w    = threadIdx.x >> 5;
  const int ln   = lane & 15;   // A/C row == B/C col
  const int hi   = lane >> 4;   // lane half -> K-pair / row-half
  const int kA   = hi << 1;     // contraction sub-index base {0,2}
  const int cb   = w * 32;

  __shared__ float vlds[L_];        // staged filter column eig_vecs[:, k]
  __shared__ float xlds[16 * D_];   // X_k tile [16 x 256]
  const unsigned vlds_lds = lds_addr32(vlds);

  const float* ub = inp + (size_t)b * L_ * D_;

  for (int pass = 0; pass < 2; ++pass) {
    const int i  = pass ? (NT_ - 1 - bx) : bx;  // balanced tile pair
    const int t0 = i * 16;

    v8f acc0 = {0.f, 0.f, 0.f, 0.f, 0.f, 0.f, 0.f, 0.f};
    v8f acc1 = {0.f, 0.f, 0.f, 0.f, 0.f, 0.f, 0.f, 0.f};

    // ---- AR term: sum_kk u[t-kk, :] @ m_u[:, :, kk]^T ----
    if (t0 >= KU_) {  // no boundary: branch-free loads
      const float* ur0 = ub + (size_t)(t0 + ln) * D_;
      for (int kk = 0; kk < KU_; ++kk) {
        const float* ur = ur0 - (size_t)kk * D_;
        for (int q = 0; q < D_ / 4; ++q) {
          const int c0 = 4 * q + kA;
          v2f a;
          a.x = ur[c0];
          a.y = ur[c0 + 1];
          v2f p0, p1;
          p0.x = m_u[((size_t)(cb + ln) * D_ + c0) * KU_ + kk];
          p0.y = m_u[((size_t)(cb + ln) * D_ + c0 + 1) * KU_ + kk];
          p1.x = m_u[((size_t)(cb + 16 + ln) * D_ + c0) * KU_ + kk];
          p1.y = m_u[((size_t)(cb + 16 + ln) * D_ + c0 + 1) * KU_ + kk];
          acc0 = wmma_f32(a, p0, acc0);
          acc1 = wmma_f32(a, p1, acc1);
        }
      }
    } else {  // tile 0 only: guarded
      for (int kk = 0; kk < KU_; ++kk) {
        const int  tr = ln - kk;
        const bool ok = (tr >= 0);
        const float* ur = ub + (size_t)(ok ? tr : 0) * D_;
        for (int q = 0; q < D_ / 4; ++q) {
          const int c0 = 4 * q + kA;
          v2f a;
          a.x = ok ? ur[c0]     : 0.f;
          a.y = ok ? ur[c0 + 1] : 0.f;
          v2f p0, p1;
          p0.x = m_u[((size_t)(cb + ln) * D_ + c0) * KU_ + kk];
          p0.y = m_u[((size_t)(cb + ln) * D_ + c0 + 1) * KU_ + kk];
          p1.x = m_u[((size_t)(cb + 16 + ln) * D_ + c0) * KU_ + kk];
          p1.y = m_u[((size_t)(cb + 16 + ln) * D_ + c0 + 1) * KU_ + kk];
          acc0 = wmma_f32(a, p0, acc0);
          acc1 = wmma_f32(a, p1, acc1);
        }
      }
    }

    // ---- Spectral term, fused per filter k ----
    for (int k = 0; k < K_; ++k) {
      __syncthreads();  // prior readers of vlds / xlds done
      // async-stage eig_vecs[0:t0+16, k] into LDS (strided gather)
      for (int idx = threadIdx.x; idx < t0 + 16; idx += 256)
        async_f32_to_lds(vlds_lds + 4u * idx,
                         eig_vecs + (size_t)idx * K_ + k);
      wait_async0();
      __syncthreads();

      // X_k[t, c] = sum_tau V_k[t - tau] u[tau, c]  (Toeplitz matmul)
      v8f x0 = {0.f, 0.f, 0.f, 0.f, 0.f, 0.f, 0.f, 0.f};
      v8f x1 = {0.f, 0.f, 0.f, 0.f, 0.f, 0.f, 0.f, 0.f};
      for (int j = 0; j < i; ++j) {  // d = i-j >= 1: branch-free
        const int base = (i - j) * 16 + ln;
        const float* urow = ub + (size_t)(16 * j) * D_ + cb + ln;
#pragma unroll
        for (int q = 0; q < 4; ++q) {
          const int tl = 4 * q + kA;
          v2f a;
          a.x = vlds[base - tl];
          a.y = vlds[base - tl - 1];
          const float* r0 = urow + (size_t)tl * D_;
          v2f u0, u1;
          u0.x = r0[0];
          u0.y = r0[D_];
          u1.x = r0[16];
          u1.y = r0[D_ + 16];
          x0 = wmma_f32(a, u0, x0);
          x1 = wmma_f32(a, u1, x1);
        }
      }
      {  // diagonal tile j == i (d = 0): triangular, guarded
        const float* urow = ub + (size_t)t0 * D_ + cb + ln;
#pragma unroll
        for (int q = 0; q < 4; ++q) {
          const int tl = 4 * q + kA;
          const int vi = ln - tl;
          v2f a;
          a.x = (vi >= 0) ? vlds[vi]     : 0.f;
          a.y = (vi >= 1) ? vlds[vi - 1] : 0.f;
          const float* r0 = urow + (size_t)tl * D_;
          v2f u0, u1;
          u0.x = r0[0];
          u0.y = r0[D_];
          u1.x = r0[16];
          u1.y = r0[D_ + 16];
          x0 = wmma_f32(a, u0, x0);
          x1 = wmma_f32(a, u1, x1);
        }
      }

      // scale by eig_vals[k]^{1/4}, publish X_k tile
      const float scale = sqrtf(sqrtf(eig_vals[k]));
#pragma unroll
      for (int r = 0; r < 8; ++r) {
        const int row = r + 8 * hi;
        xlds[row * D_ + cb + ln]      = x0[r] * scale;
        xlds[row * D_ + cb + 16 + ln] = x1[r] * scale;
      }
      __syncthreads();

      // delta += X_k @ m_phi[k*D:(k+1)*D, :]   (A from LDS as b64 loads)
      const float* phik = m_phi + (size_t)k * D_ * D_;
      for (int q = 0; q < D_ / 4; ++q) {
        const int c0 = 4 * q + kA;  // even -> 8B aligned
        const v2f a = *(const v2f*)&xlds[ln * D_ + c0];
        const float* pr = phik + (size_t)c0 * D_ + cb + ln;
        v2f p0, p1;
        p0.x = pr[0];
        p0.y = pr[D_];
        p1.x = pr[16];
        p1.y = pr[D_ + 16];
        acc0 = wmma_f32(a, p0, acc0);
        acc1 = wmma_f32(a, p1, acc1);
      }
    }  // k

    // ---- store delta tile ----
    float* drow = delta + ((size_t)b * L_ + t0) * D_;
#pragma unroll
    for (int r = 0; r < 8; ++r) {
      const int row = r + 8 * hi;
      drow[(size_t)row * D_ + cb + ln]      = acc0[r];
      drow[(size_t)row * D_ + cb + 16 + ln] = acc1[r];
    }
  }  // pass
}

// ---------------------------------------------------------------------------
// Kernel 2: y_t = M0 @ y_{t-1} + M1 @ y_{t-2} + delta_t  (sequential scan)
// grid = B, block = 512: thread (o = tid>>1, p = tid&1); p selects the
// (M0, y_{t-1}) or (M1, y_{t-2}) half. 192 of 256 weights register-cached.
// ---------------------------------------------------------------------------
#define MR_ 48  // float4s of M kept in registers (192 floats; ~230 VGPR total)

__global__ __launch_bounds__(512) void stu_scan_kernel(
    const float* __restrict__ delta,  // [B, L, D]
    const float* __restrict__ m_y,    // [D, KY, D]
    float* __restrict__ out) {        // [B, L, D]
  const int b = blockIdx.x;
  const int o = threadIdx.x >> 1;
  const int p = threadIdx.x & 1;
  __shared__ float ybuf[2][D_];
  ((float*)ybuf)[threadIdx.x] = 0.f;  // 512 threads cover 2*256 exactly
  __syncthreads();

  // this thread's half-row: m_y[o, p, :]
  const float4* Mrow = (const float4*)(m_y + (size_t)o * (KY_ * D_) +
                                       (size_t)p * D_);
  float4 mreg[MR_];
#pragma unroll
  for (int j = 0; j < MR_; ++j) mreg[j] = Mrow[j];

  const float* drow = delta + (size_t)b * L_ * D_;
  float* orow = out + (size_t)b * L_ * D_;

  for (int t = 0; t < L_; ++t) {
    // p=0 -> y_{t-1} (slot (t+1)&1), p=1 -> y_{t-2} (slot t&1)
    const float4* yv = (const float4*)ybuf[(t + 1 + p) & 1];
    float s = 0.f;
#pragma unroll
    for (int j = 0; j < MR_; ++j) {
      const float4 m4 = mreg[j];
      const float4 y4 = yv[j];
      s += m4.x * y4.x + m4.y * y4.y + m4.z * y4.z + m4.w * y4.w;
    }
#pragma unroll
    for (int j = MR_; j < D_ / 4; ++j) {  // streamed tail (L2-resident)
      const float4 m4 = Mrow[j];
      const float4 y4 = yv[j];
      s += m4.x * y4.x + m4.y * y4.y + m4.z * y4.z + m4.w * y4.w;
    }
    s += __shfl_xor(s, 1, 32);  // combine the two halves (adjacent lanes)
    __syncthreads();            // all reads of y buffers complete
    if (p == 0) {
      const float v = s + drow[(size_t)t * D_ + o];
      ybuf[t & 1][o] = v;       // y_t overwrites y_{t-2} slot
      orow[(size_t)t * D_ + o] = v;
    }
    __syncthreads();            // y_t visible before next step
  }
}

// ---------------------------------------------------------------------------
extern "C" void kernel_launch(void* const* d_in, const int* in_sizes, int n_in,
                              void* d_out, int out_size, void* d_ws,
                              size_t ws_size, hipStream_t stream) {
  const float* inp      = (const float*)d_in[0];  // [B, L, D]
  const float* m_y      = (const float*)d_in[1];  // [D, KY, D]
  const float* m_u      = (const float*)d_in[2];  // [D, D, KU]
  const float* m_phi    = (const float*)d_in[3];  // [K*D, D]
  const float* eig_vals = (const float*)d_in[4];  // [K]
  const float* eig_vecs = (const float*)d_in[5];  // [L, K]
  float* out   = (float*)d_out;
  float* delta = (float*)d_ws;                    // [B, L, D] f32 = 8 MB

  dim3 grid1(NT_ / 2, B_);
  stu_delta_kernel<<<grid1, 256, 0, stream>>>(inp, m_u, m_phi, eig_vals,
                                              eig_vecs, delta);
  stu_scan_kernel<<<dim3(B_), 512, 0, stream>>>(delta, m_y, out);
}